// GCNLayer_1065151889944
// MI455X (gfx1250) — compile-verified
//
#include <hip/hip_runtime.h>

#define N_NODES 100000
#define N_EDGES 1200000
#define D 64
#define ROW_TILES (N_NODES / 16)   // 6250, exact

typedef __attribute__((ext_vector_type(2))) float v2f;
typedef __attribute__((ext_vector_type(8))) float v8f;

// ---------------- kernel 1: zero output + degree arrays ----------------
__global__ void zero_kernel(float* __restrict__ out, float* __restrict__ deg,
                            int n_out, int n_deg) {
    int g = blockIdx.x * blockDim.x + threadIdx.x;
    if (g < n_out) out[g] = 0.0f;
    if (g < n_deg) deg[g] = 0.0f;
}

// ---------------- kernel 2: degree accumulation ----------------
__global__ void degree_kernel(const int* __restrict__ src, const int* __restrict__ dst,
                              float* __restrict__ deg_out, float* __restrict__ deg_in) {
    int e = blockIdx.x * blockDim.x + threadIdx.x;
    if (e < N_EDGES) {
        atomicAdd(&deg_out[src[e]], 1.0f);
        atomicAdd(&deg_in[dst[e]], 1.0f);
    }
}

// ---------------- kernel 3: h = (x @ W) * norm_src, fp32 WMMA ----------------
// One wave computes a 16-row x 64-col tile of h.
// A fragment (16x4 f32): lane<16 -> {K=0,K=1}, lane>=16 -> {K=2,K=3}, M = lane%16
// B fragment (4x16 f32): v[0] holds row k0+2*half, v[1] holds row k0+2*half+1, N = lane%16
// C/D (16x16 f32): VGPR r, lane l -> M = r + 8*(l/16), N = l%16
__global__ __launch_bounds__(256) void gemm_scale_kernel(
    const float* __restrict__ x, const float* __restrict__ W,
    const float* __restrict__ deg_out, float* __restrict__ h) {
    const int lane = threadIdx.x & 31;
    const int wave = threadIdx.x >> 5;
    const int tile = blockIdx.x * 8 + wave;   // wave-uniform
    if (tile >= ROW_TILES) return;            // whole wave exits together (EXEC all-ones for WMMA)

    const int row0 = tile * 16;
    const int m    = lane & 15;
    const int half = lane >> 4;

    v8f acc[4] = {v8f{}, v8f{}, v8f{}, v8f{}};

    const float* __restrict__ xrow = x + (row0 + m) * D;

#pragma unroll
    for (int k0 = 0; k0 < D; k0 += 4) {
        v2f a;
        a.x = xrow[k0 + 2 * half];
        a.y = xrow[k0 + 2 * half + 1];
        const float* __restrict__ wk = W + (k0 + 2 * half) * D;  // W row (k0 + 2*half)
#pragma unroll
        for (int j = 0; j < 4; ++j) {
            v2f b;
            b.x = wk[j * 16 + m];        // row k0+2*half
            b.y = wk[D + j * 16 + m];    // row k0+2*half+1
            acc[j] = __builtin_amdgcn_wmma_f32_16x16x4_f32(
                /*neg_a=*/false, a, /*neg_b=*/false, b,
                /*c_mod=*/(short)0, acc[j], /*reuse_a=*/false, /*reuse_b=*/false);
        }
    }

    // per-row source normalization: rsqrt(max(deg_out,1))
    float ns[8];
#pragma unroll
    for (int r = 0; r < 8; ++r) {
        int row = row0 + half * 8 + r;
        ns[r] = rsqrtf(fmaxf(deg_out[row], 1.0f));
    }
#pragma unroll
    for (int j = 0; j < 4; ++j) {
#pragma unroll
        for (int r = 0; r < 8; ++r) {
            int row = row0 + half * 8 + r;
            h[row * D + j * 16 + m] = acc[j][r] * ns[r];
        }
    }
}

// ---------------- kernel 4: edge scatter-add ----------------
// thread = (edge, 16-byte column chunk): float4 gather from L2-resident h,
// 4 coalesced f32 atomics into out[dst].
__global__ void scatter_kernel(const int* __restrict__ src, const int* __restrict__ dst,
                               const float* __restrict__ h, float* __restrict__ out) {
    int g = blockIdx.x * blockDim.x + threadIdx.x;
    int e  = g >> 4;
    int c4 = (g & 15) * 4;
    if (e < N_EDGES) {
        int s = src[e];
        int d = dst[e];
        const float4 v = *(const float4*)(h + s * D + c4);
        float* op = out + d * D + c4;
        atomicAdd(op + 0, v.x);
        atomicAdd(op + 1, v.y);
        atomicAdd(op + 2, v.z);
        atomicAdd(op + 3, v.w);
    }
}

// ---------------- kernel 5: out = relu(out * norm_dst + b) ----------------
__global__ void finalize_kernel(float* __restrict__ out, const float* __restrict__ deg_in,
                                const float* __restrict__ bias) {
    int g = blockIdx.x * blockDim.x + threadIdx.x;
    if (g < N_NODES * D) {
        int row = g >> 6;
        float nd = rsqrtf(fmaxf(deg_in[row], 1.0f));
        float v = fmaf(out[g], nd, bias[g & (D - 1)]);
        out[g] = fmaxf(v, 0.0f);
    }
}

extern "C" void kernel_launch(void* const* d_in, const int* in_sizes, int n_in,
                              void* d_out, int out_size, void* d_ws, size_t ws_size,
                              hipStream_t stream) {
    const float* x   = (const float*)d_in[0];   // [N, 64]
    const float* W   = (const float*)d_in[1];   // [64, 64]
    const float* b   = (const float*)d_in[2];   // [64]
    const int*   src = (const int*)d_in[3];     // [E]
    const int*   dst = (const int*)d_in[4];     // [E]
    float* out = (float*)d_out;                 // [N, 64]

    // workspace layout: deg_out[N] | deg_in[N] | h[N*64]   (~26.4 MB)
    float* ws      = (float*)d_ws;
    float* deg_out = ws;
    float* deg_in  = ws + N_NODES;
    float* h       = ws + 2 * N_NODES;

    const int THREADS = 256;
    const int n_out = N_NODES * D;
    const int n_deg = 2 * N_NODES;

    // 1) zero out + degrees
    zero_kernel<<<(n_out + THREADS - 1) / THREADS, THREADS, 0, stream>>>(out, deg_out, n_out, n_deg);

    // 2) degree accumulation
    degree_kernel<<<(N_EDGES + THREADS - 1) / THREADS, THREADS, 0, stream>>>(src, dst, deg_out, deg_in);

    // 3) h = (x @ W) * norm_src  via fp32 WMMA; 8 waves (= 8 row-tiles) per block
    int gemm_blocks = (ROW_TILES + 7) / 8;
    gemm_scale_kernel<<<gemm_blocks, THREADS, 0, stream>>>(x, W, deg_out, h);

    // 4) edge scatter: E * 16 threads (float4 chunks)
    long long scatter_threads = (long long)N_EDGES * 16;
    scatter_kernel<<<(int)((scatter_threads + THREADS - 1) / THREADS), THREADS, 0, stream>>>(src, dst, h, out);

    // 5) finalize: normalize by dst degree, add bias, relu
    finalize_kernel<<<(n_out + THREADS - 1) / THREADS, THREADS, 0, stream>>>(out, deg_in, b);
}